// CriticNetwork_35081292874063
// MI455X (gfx1250) — compile-verified
//
#include <hip/hip_runtime.h>

typedef float f32x2 __attribute__((ext_vector_type(2)));
typedef float f32x8 __attribute__((ext_vector_type(8)));

#define N_NODES 50000
#define N_EDGES 800000
#define N_AGENTS 16384
#define IN_DIM 128
#define HID 256
#define FC1 1024
#define FC2 512
#define N_ACT 64

// ---------------------------------------------------------------------------
// CDNA5 async global->LDS copy (ASYNCcnt-tracked), inline asm for portability
// across ROCm7.2 / amdgpu-toolchain builtin arity differences.
// ---------------------------------------------------------------------------
__device__ __forceinline__ void async_ld_b128(unsigned lds_byte_addr, const void* gaddr) {
    asm volatile("global_load_async_to_lds_b128 %0, %1, off"
                 :: "v"(lds_byte_addr), "v"((unsigned long long)(uintptr_t)gaddr)
                 : "memory");
}
__device__ __forceinline__ void wait_async0() {
    asm volatile("s_wait_asynccnt 0x0" ::: "memory");
}

// ---------------------------------------------------------------------------
// Degree kernels
// ---------------------------------------------------------------------------
__global__ __launch_bounds__(256) void deg_init(float* __restrict__ deg, int n) {
    int i = blockIdx.x * blockDim.x + threadIdx.x;
    if (i < n) deg[i] = 1.0f;  // self loop
}

__global__ __launch_bounds__(256) void deg_accum(const int* __restrict__ dst,
                                                 float* __restrict__ deg, int e) {
    int i = blockIdx.x * blockDim.x + threadIdx.x;
    if (i < e) unsafeAtomicAdd(&deg[dst[i]], 1.0f);
}

__global__ __launch_bounds__(256) void deg_rsqrt(float* __restrict__ deg, int n) {
    int i = blockIdx.x * blockDim.x + threadIdx.x;
    if (i < n) deg[i] = rsqrtf(deg[i]);  // deg >= 1 always
}

// ---------------------------------------------------------------------------
// Weight transpose: BT[n*K + k] = B[k*N + n]  (weights are tiny, L2 resident)
// ---------------------------------------------------------------------------
__global__ __launch_bounds__(256)
void transpose_kn(const float* __restrict__ B, float* __restrict__ BT, int K, int N) {
    int i = blockIdx.x * blockDim.x + threadIdx.x;
    if (i < K * N) {
        int k = i / N, n = i % N;
        BT[(size_t)n * K + k] = B[i];
    }
}

// ---------------------------------------------------------------------------
// FP32 WMMA GEMM, LDS-staged B:  C[M,N] = A[M,K] @ B[K,N] (+bias[N])
//   - BT is B transposed [N][K] so a B fragment is one contiguous 8B load
//   - block = 8 waves -> 128 rows x 64 cols; all waves share one B stripe
//   - B stripe staged in 64xKC chunks via async-to-LDS, double-buffered
// Fragment layouts (wave32, \S7.12.2):
//   A 16x4 f32 : lane h=lane>>4, r=lane&15 -> {A[r][k+2h], A[r][k+2h+1]}
//   B 4x16 f32 : c=lane&15              -> {B[k+2h][c],  B[k+2h+1][c]}
//   C/D 16x16  : VGPR r -> row r (lanes 0-15) / r+8 (lanes 16-31)
// ---------------------------------------------------------------------------
template <int K, bool HAS_BIAS>
__global__ __launch_bounds__(256)
void gemm_wmma_stage(const float* __restrict__ A, const float* __restrict__ BT,
                     const float* __restrict__ bias, float* __restrict__ C,
                     int M, int N) {
    constexpr int KC = 32;
    constexpr int NCHUNK = K / KC;
    __shared__ __align__(16) float lb[2][64 * KC];

    const int tid  = threadIdx.x;
    const int lane = tid & 31;
    const int wv   = tid >> 5;
    const int ntiles = N >> 6;
    const int bm   = blockIdx.x / ntiles;
    const int tn   = blockIdx.x % ntiles;
    const int half = lane >> 4;
    const int l15  = lane & 15;
    const int tm16 = bm * 128 + wv * 16;
    const bool active = (tm16 < M);

    // Staging plan: 64 x KC floats = 8KB = 512 x 16B; 2 items / thread.
    //   item t: c = t>>3 (0..63), kq = (t&7)*4
    const int c0  = tid >> 3;
    const int kq0 = (tid & 7) << 2;
    const float*   gB0   = BT + (size_t)(tn * 64 + c0)      * K + kq0;
    const float*   gB1   = BT + (size_t)(tn * 64 + c0 + 32) * K + kq0;
    const unsigned loff0 = (unsigned)((c0 * KC + kq0) * 4);
    const unsigned loff1 = (unsigned)(((c0 + 32) * KC + kq0) * 4);
    const unsigned lbase0 = (unsigned)(uintptr_t)&lb[0][0];
    const unsigned lbase1 = (unsigned)(uintptr_t)&lb[1][0];

    const float* Ap = A + (size_t)(active ? (tm16 + l15) : 0) * K + 2 * half;

    f32x8 acc[4] = {};

    // prime buffer 0 with chunk 0
    async_ld_b128(lbase0 + loff0, gB0);
    async_ld_b128(lbase0 + loff1, gB1);

    for (int ch = 0; ch < NCHUNK; ++ch) {
        wait_async0();
        __syncthreads();  // chunk ch visible to all waves
        const int buf = ch & 1;
        if (ch + 1 < NCHUNK) {  // kick off next chunk's DMA into other buffer
            const unsigned lb2 = buf ? lbase0 : lbase1;
            const int kn = (ch + 1) * KC;
            async_ld_b128(lb2 + loff0, gB0 + kn);
            async_ld_b128(lb2 + loff1, gB1 + kn);
        }
        if (active) {
            const float* lbuf = &lb[buf][0];
            const int k0 = ch * KC;
#pragma unroll
            for (int kk = 0; kk < KC; kk += 4) {
                f32x2 a = *(const f32x2*)(Ap + k0 + kk);
#pragma unroll
                for (int j = 0; j < 4; ++j) {
                    f32x2 b = *(const f32x2*)(lbuf + (j * 16 + l15) * KC + kk + 2 * half);
                    acc[j] = __builtin_amdgcn_wmma_f32_16x16x4_f32(
                        false, a, false, b, (short)0, acc[j], false, false);
                }
            }
        }
        // no trailing barrier needed: restage of this buffer happens after the
        // *next* iteration's leading barrier, which orders all consumption.
    }

    if (active) {
#pragma unroll
        for (int j = 0; j < 4; ++j) {
            const int col  = tn * 64 + j * 16 + l15;
            const float bv = HAS_BIAS ? bias[col] : 0.0f;
            float* Cp = C + (size_t)(tm16 + 8 * half) * N + col;
#pragma unroll
            for (int r = 0; r < 8; ++r)
                Cp[(size_t)r * N] = acc[j][r] + bv;
        }
    }
}

// ---------------------------------------------------------------------------
// GCN aggregation
// ---------------------------------------------------------------------------
__global__ __launch_bounds__(256)
void agg_init(const float* __restrict__ h, const float* __restrict__ dinv,
              float* __restrict__ agg, int n) {  // n = N_NODES*HID
    int i = blockIdx.x * blockDim.x + threadIdx.x;
    if (i < n) {
        int node = i >> 8;  // HID == 256
        float d = dinv[node];
        agg[i] = h[i] * d * d;  // self-loop contribution
    }
}

__global__ __launch_bounds__(256)
void edge_scatter(const int* __restrict__ src, const int* __restrict__ dst,
                  const float* __restrict__ dinv, const float* __restrict__ h,
                  float* __restrict__ agg, int e) {
    unsigned tid = blockIdx.x * blockDim.x + threadIdx.x;
    int edge = tid >> 6;  // 64 threads per edge, 4 floats each
    if (edge >= e) return;
    int g = (tid & 63) << 2;
    int s = src[edge], d = dst[edge];
    float nrm = dinv[s] * dinv[d];
    const float4 hv = *(const float4*)(h + (size_t)s * HID + g);
    float* ap = agg + (size_t)d * HID + g;
    unsafeAtomicAdd(ap + 0, hv.x * nrm);
    unsafeAtomicAdd(ap + 1, hv.y * nrm);
    unsafeAtomicAdd(ap + 2, hv.z * nrm);
    unsafeAtomicAdd(ap + 3, hv.w * nrm);
}

__global__ __launch_bounds__(256)
void gather_relu(const float* __restrict__ agg, const int* __restrict__ idx,
                 const float* __restrict__ bgcn, float* __restrict__ z, int n) {
    int i = blockIdx.x * blockDim.x + threadIdx.x;
    if (i < n) {
        int a = i >> 8, j = i & 255;
        float v = agg[(size_t)idx[a] * HID + j] + bgcn[j];
        z[i] = fmaxf(v, 0.0f);
    }
}

// ---------------------------------------------------------------------------
// Row LayerNorm (in place) with optional ReLU. One block (256 thr) per row.
// ---------------------------------------------------------------------------
__global__ __launch_bounds__(256)
void layernorm_rows(float* __restrict__ X, const float* __restrict__ g,
                    const float* __restrict__ beta, int W, int relu) {
    const int tid = threadIdx.x;
    float* Xr = X + (size_t)blockIdx.x * W;
    __shared__ float sh1[256], sh2[256];
    float s = 0.f, s2 = 0.f;
    for (int j = tid; j < W; j += 256) { float v = Xr[j]; s += v; s2 += v * v; }
    sh1[tid] = s; sh2[tid] = s2;
    __syncthreads();
    for (int off = 128; off > 0; off >>= 1) {
        if (tid < off) { sh1[tid] += sh1[tid + off]; sh2[tid] += sh2[tid + off]; }
        __syncthreads();
    }
    const float mu  = sh1[0] / W;
    const float var = sh2[0] / W - mu * mu;
    const float ri  = rsqrtf(var + 1e-5f);
    for (int j = tid; j < W; j += 256) {
        float v = (Xr[j] - mu) * ri * g[j] + beta[j];
        Xr[j] = relu ? fmaxf(v, 0.f) : v;
    }
}

// ---------------------------------------------------------------------------
// Fused tail: out[row] = relu(LN(t2_row; g2,b2) + a_row) . Wq + bq
// ---------------------------------------------------------------------------
__global__ __launch_bounds__(256)
void final_fused(const float* __restrict__ T, const float* __restrict__ Ab,
                 const float* __restrict__ g, const float* __restrict__ beta,
                 const float* __restrict__ Wq, const float* __restrict__ bq,
                 float* __restrict__ out) {
    const int tid = threadIdx.x;
    const float* Tr = T  + (size_t)blockIdx.x * FC2;
    const float* Ar = Ab + (size_t)blockIdx.x * FC2;
    __shared__ float sh1[256], sh2[256];
    const float v0 = Tr[tid], v1 = Tr[tid + 256];
    sh1[tid] = v0 + v1;
    sh2[tid] = v0 * v0 + v1 * v1;
    __syncthreads();
    for (int off = 128; off > 0; off >>= 1) {
        if (tid < off) { sh1[tid] += sh1[tid + off]; sh2[tid] += sh2[tid + off]; }
        __syncthreads();
    }
    const float mu  = sh1[0] / FC2;
    const float var = sh2[0] / FC2 - mu * mu;
    const float ri  = rsqrtf(var + 1e-5f);
    const float d0 = fmaxf((v0 - mu) * ri * g[tid]       + beta[tid]       + Ar[tid],       0.f) * Wq[tid];
    const float d1 = fmaxf((v1 - mu) * ri * g[tid + 256] + beta[tid + 256] + Ar[tid + 256], 0.f) * Wq[tid + 256];
    __syncthreads();
    sh1[tid] = d0 + d1;
    __syncthreads();
    for (int off = 128; off > 0; off >>= 1) {
        if (tid < off) sh1[tid] += sh1[tid + off];
        __syncthreads();
    }
    if (tid == 0) out[blockIdx.x] = sh1[0] + bq[0];
}

// ---------------------------------------------------------------------------
extern "C" void kernel_launch(void* const* d_in, const int* in_sizes, int n_in,
                              void* d_out, int out_size, void* d_ws, size_t ws_size,
                              hipStream_t stream) {
    (void)in_sizes; (void)n_in; (void)out_size; (void)ws_size;
    const float* x      = (const float*)d_in[0];
    const float* action = (const float*)d_in[1];
    const int*   eidx   = (const int*)d_in[2];
    const int*   aidx   = (const int*)d_in[3];
    const float* Wg     = (const float*)d_in[4];
    const float* bg     = (const float*)d_in[5];
    const float* W1     = (const float*)d_in[6];
    const float* b1     = (const float*)d_in[7];
    const float* g1     = (const float*)d_in[8];
    const float* be1    = (const float*)d_in[9];
    const float* W2     = (const float*)d_in[10];
    const float* b2     = (const float*)d_in[11];
    const float* g2     = (const float*)d_in[12];
    const float* be2    = (const float*)d_in[13];
    const float* Wa     = (const float*)d_in[14];
    const float* ba     = (const float*)d_in[15];
    const float* Wq     = (const float*)d_in[16];
    const float* bq     = (const float*)d_in[17];
    float* out = (float*)d_out;

    const int* src = eidx;
    const int* dst = eidx + N_EDGES;

    // workspace layout (floats); abuf overlays h (h dead after edge_scatter)
    float* ws   = (float*)d_ws;
    float* deg  = ws;                                   // N_NODES
    float* h    = ws + 50048;                           // N_NODES*HID
    float* agg  = h   + (size_t)N_NODES * HID;          // N_NODES*HID
    float* z    = agg + (size_t)N_NODES * HID;          // N_AGENTS*HID
    float* s1   = z   + (size_t)N_AGENTS * HID;         // N_AGENTS*FC1
    float* t2   = s1  + (size_t)N_AGENTS * FC1;         // N_AGENTS*FC2
    float* wgT  = t2  + (size_t)N_AGENTS * FC2;         // HID*IN_DIM
    float* w1T  = wgT + (size_t)HID * IN_DIM;           // FC1*HID
    float* w2T  = w1T + (size_t)FC1 * HID;              // FC2*FC1
    float* waT  = w2T + (size_t)FC2 * FC1;              // FC2*N_ACT
    float* ab   = h;                                    // N_AGENTS*FC2 (overlay)

    // 0) transpose the (tiny) weights so B fragments are contiguous 8B loads
    transpose_kn<<<(IN_DIM * HID + 255) / 256, 256, 0, stream>>>(Wg, wgT, IN_DIM, HID);
    transpose_kn<<<(HID * FC1 + 255) / 256, 256, 0, stream>>>(W1, w1T, HID, FC1);
    transpose_kn<<<(FC1 * FC2 + 255) / 256, 256, 0, stream>>>(W2, w2T, FC1, FC2);
    transpose_kn<<<(N_ACT * FC2 + 255) / 256, 256, 0, stream>>>(Wa, waT, N_ACT, FC2);

    // 1) symmetric degree normalization
    deg_init <<<(N_NODES + 255) / 256, 256, 0, stream>>>(deg, N_NODES);
    deg_accum<<<(N_EDGES + 255) / 256, 256, 0, stream>>>(dst, deg, N_EDGES);
    deg_rsqrt<<<(N_NODES + 255) / 256, 256, 0, stream>>>(deg, N_NODES);

    // 2) h = x @ W_gcn   [50000,128] x [128,256]
    {
        int blocks = ((N_NODES + 127) / 128) * (HID / 64);
        gemm_wmma_stage<IN_DIM, false><<<blocks, 256, 0, stream>>>(x, wgT, nullptr, h,
                                                                   N_NODES, HID);
    }

    // 3) aggregation: self-loop init + edge scatter
    {
        int n = N_NODES * HID;
        agg_init<<<(n + 255) / 256, 256, 0, stream>>>(h, deg, agg, n);
        long long th = (long long)N_EDGES * 64;
        edge_scatter<<<(int)((th + 255) / 256), 256, 0, stream>>>(src, dst, deg, h,
                                                                  agg, N_EDGES);
    }

    // 4) z = relu(agg[agent_idx] + b_gcn)
    {
        int n = N_AGENTS * HID;
        gather_relu<<<(n + 255) / 256, 256, 0, stream>>>(agg, aidx, bg, z, n);
    }

    // 5) s1 = relu(LN(z @ W1 + b1))
    {
        int blocks = (N_AGENTS / 128) * (FC1 / 64);
        gemm_wmma_stage<HID, true><<<blocks, 256, 0, stream>>>(z, w1T, b1, s1,
                                                               N_AGENTS, FC1);
        layernorm_rows<<<N_AGENTS, 256, 0, stream>>>(s1, g1, be1, FC1, 1);
    }

    // 6) ab = action @ Wa + ba   (overlays h, which is dead now)
    {
        int blocks = (N_AGENTS / 128) * (FC2 / 64);
        gemm_wmma_stage<N_ACT, true><<<blocks, 256, 0, stream>>>(action, waT, ba, ab,
                                                                 N_AGENTS, FC2);
    }

    // 7) t2 = s1 @ W2 + b2
    {
        int blocks = (N_AGENTS / 128) * (FC2 / 64);
        gemm_wmma_stage<FC1, true><<<blocks, 256, 0, stream>>>(s1, w2T, b2, t2,
                                                               N_AGENTS, FC2);
    }

    // 8) out = relu(LN(t2; g2,b2) + ab) @ Wq + bq
    final_fused<<<N_AGENTS, 256, 0, stream>>>(t2, ab, g2, be2, Wq, bq, out);
}